// PrototypeLayer_47699906789485
// MI455X (gfx1250) — compile-verified
//
#include <hip/hip_runtime.h>
#include <math.h>

typedef float v2f __attribute__((ext_vector_type(2)));
typedef float v8f __attribute__((ext_vector_type(8)));

#define NB    256
#define NP    200
#define ND    1024
#define NSUB  16
#define WAVES_PER_BLOCK 8
#define SWEEPS 6

__launch_bounds__(256, 2)
__global__ void proto_grassmann_kernel(const float* __restrict__ xs,
                                       const float* __restrict__ xprotos,
                                       const float* __restrict__ rel,
                                       float* __restrict__ out)
{
    // 64 KB: X_b staged for GEMM, then reused as per-wave 16x16 M tiles.
    __shared__ float sX[ND * NSUB];

    const int tid  = threadIdx.x;
    const int lane = tid & 31;
    const int wave = tid >> 5;
    const int half = lane >> 4;   // K-split half of the wave (WMMA f32 A/B layout)
    const int l    = lane & 15;

    const int tile = blockIdx.x * WAVES_PER_BLOCK + wave;   // [0, B*P)
    const int b    = (blockIdx.x * WAVES_PER_BLOCK) / NP;   // same for all 8 waves
    const int p    = tile % NP;

    // ---- cooperative stage of X_b (D x d row-major) into LDS, 128b loads ----
    {
        const float4* __restrict__ src = (const float4*)(xs + (size_t)b * ND * NSUB);
        float4* dst = (float4*)sX;
        #pragma unroll 4
        for (int i = tid; i < (ND * NSUB) / 4; i += 256)
            dst[i] = src[i];
    }
    __syncthreads();

    // ---- M = X_b^T * P_p : 16x16 accum over K=1024 via v_wmma_f32_16x16x4_f32 ----
    // A (16x4 f32): lanes 0-15 rows M=0..15 hold K={k0,k0+1}; lanes 16-31 hold K={k0+2,k0+3}
    // B (4x16 f32): v0 = row K=k0 (lanes 0-15) / K=k0+2 (lanes 16-31); v1 = odd rows
    const float* __restrict__ Bp = xprotos + (size_t)p * ND * NSUB;
    v8f acc = {};
    #pragma unroll 8
    for (int k0 = 0; k0 < ND; k0 += 4) {
        const int kr = k0 + 2 * half;
        v2f a, bf;
        a.x  = sX[kr * NSUB + l];
        a.y  = sX[(kr + 1) * NSUB + l];
        bf.x = Bp[kr * NSUB + l];
        bf.y = Bp[(kr + 1) * NSUB + l];
        acc = __builtin_amdgcn_wmma_f32_16x16x4_f32(
            /*neg_a=*/false, a, /*neg_b=*/false, bf,
            /*c_mod=*/(short)0, acc, /*reuse_a=*/false, /*reuse_b=*/false);
    }

    __syncthreads();  // all waves finished reading sX; safe to reuse as M tiles

    // C layout: VGPR r -> row (r + 8*half), col = l. Repack to column-per-lane via LDS.
    float* sM = sX + wave * 256;
    #pragma unroll
    for (int r = 0; r < 8; ++r)
        sM[(r + 8 * half) * 16 + l] = acc[r];

    // ---- one-sided Jacobi SVD: lane l owns column l of M and of V (halves mirror) ----
    float m[16], v[16];
    #pragma unroll
    for (int i = 0; i < 16; ++i) {
        m[i] = sM[i * 16 + l];
        v[i] = (i == l) ? 1.0f : 0.0f;
    }

    for (int sweep = 0; sweep < SWEEPS; ++sweep) {
        #pragma unroll 1
        for (int k = 1; k < 16; ++k) {       // xor tournament: covers all 120 pairs
            const int j = l ^ k;
            float o[16], ov[16];
            float app = 0.f, apq = 0.f;
            #pragma unroll
            for (int i = 0; i < 16; ++i) {
                o[i]  = __shfl_xor(m[i], k, 32);
                ov[i] = __shfl_xor(v[i], k, 32);
                app += m[i] * m[i];
                apq += m[i] * o[i];
            }
            const float aqq  = __shfl_xor(app, k, 32);
            const bool first = l < j;                 // this lane plays role of column p
            const float a_p = first ? app : aqq;
            const float a_q = first ? aqq : app;
            float c = 1.f, s = 0.f;
            if (fabsf(apq) > 1e-12f) {
                const float tau = (a_q - a_p) / (2.0f * apq);
                const float t   = copysignf(1.0f / (fabsf(tau) + sqrtf(1.0f + tau * tau)), tau);
                c = rsqrtf(1.0f + t * t);
                s = c * t;
            }
            #pragma unroll
            for (int i = 0; i < 16; ++i) {
                m[i] = first ? (c * m[i] - s * o[i]) : (s * o[i] + c * m[i]);
                v[i] = first ? (c * v[i] - s * ov[i]) : (s * ov[i] + c * v[i]);
            }
        }
    }

    // ---- singular value, squared principal angle, weighted reduction ----
    float nrm2 = 0.f;
    #pragma unroll
    for (int i = 0; i < 16; ++i) nrm2 += m[i] * m[i];
    float S = sqrtf(nrm2);
    S = fminf(fmaxf(S, 0.0f), 1.0f - 1e-7f);
    const float th = acosf(S);
    float contrib = rel[l] * th * th;
    #pragma unroll
    for (int off = 1; off < 16; off <<= 1)
        contrib += __shfl_xor(contrib, off, 32);

    if (lane == 0)
        out[tile] = contrib;                         // distance (B,P)

    // ---- Qw = V : lane l writes column l (rows strided by 16) ----
    float* __restrict__ Qw = out + (size_t)NB * NP + (size_t)tile * 256;
    if (half == 0) {
        #pragma unroll
        for (int i = 0; i < 16; ++i)
            Qw[i * 16 + l] = v[i];
    }
}

extern "C" void kernel_launch(void* const* d_in, const int* in_sizes, int n_in,
                              void* d_out, int out_size, void* d_ws, size_t ws_size,
                              hipStream_t stream) {
    const float* xs  = (const float*)d_in[0];   // (B, D, d) f32
    const float* xp  = (const float*)d_in[1];   // (P, D, d) f32
    const float* rel = (const float*)d_in[2];   // (1, d)    f32
    float* out = (float*)d_out;                 // dist (B,P) ++ Qw (B,P,d,d)

    dim3 grid((NB * NP) / WAVES_PER_BLOCK);     // 6400 blocks
    dim3 block(256);                            // 8 waves, one (b,p) tile each
    hipLaunchKernelGGL(proto_grassmann_kernel, grid, block, 0, stream,
                       xs, xp, rel, out);
}